// CRF_89300960018486
// MI455X (gfx1250) — compile-verified
//
#include <hip/hip_runtime.h>
#include <hip/hip_bf16.h>
#include <stdint.h>

#define NTAGS 256
#define SEQ_T 512
#define TAG_SOS 0
#define TAG_EOS 1

typedef unsigned int v4u __attribute__((ext_vector_type(4)));
typedef int          v4i __attribute__((ext_vector_type(4)));
typedef int          v8i __attribute__((ext_vector_type(8)));

// ---------- CDNA5 async copy (global -> LDS), ASYNCcnt tracked ----------
__device__ __forceinline__ void async_cp_f32(const float* g, float* l) {
#if __has_builtin(__builtin_amdgcn_global_load_async_to_lds_b32)
  __builtin_amdgcn_global_load_async_to_lds_b32(
      (__attribute__((address_space(1))) int*)(g),
      (__attribute__((address_space(3))) int*)(l),
      0, 0);
#else
  *l = *g;  // synchronous fallback
#endif
}

__device__ __forceinline__ void async_wait0() {
#if __has_builtin(__builtin_amdgcn_s_wait_asynccnt)
  __builtin_amdgcn_s_wait_asynccnt(0);
#else
  asm volatile("s_wait_asynccnt 0" ::: "memory");
#endif
  asm volatile("" ::: "memory");  // keep LDS reads after the wait
}

__device__ __forceinline__ void tensor_wait0() {
#if __has_builtin(__builtin_amdgcn_s_wait_tensorcnt)
  __builtin_amdgcn_s_wait_tensorcnt(0);
#else
  asm volatile("s_wait_tensorcnt 0" ::: "memory");
#endif
  asm volatile("" ::: "memory");
}

// ---------- TDM: load the full 256x256 f32 transitions tile into LDS ----------
// D# per CDNA5 ISA ch.8: group0 = {count/flags, lds_addr, global_addr, type=2},
// group1 = {mask/data_size, dims, tile dims, strides}. 2D tile == whole tensor.
#define HAVE_TDM (__has_builtin(__builtin_amdgcn_tensor_load_to_lds_d2) || \
                  __has_builtin(__builtin_amdgcn_tensor_load_to_lds))

__device__ __forceinline__ void tdm_load_trans(const float* gsrc, float* ldst) {
#if HAVE_TDM
  unsigned lds_off = (unsigned)(unsigned long long)
      (__attribute__((address_space(3))) void*)(ldst);
  unsigned long long ga = (unsigned long long)(uintptr_t)gsrc;

  v4u g0;
  g0[0] = 1u;                                            // count=1, user mode
  g0[1] = lds_off;                                       // lds_addr (bytes)
  g0[2] = (unsigned)(ga & 0xFFFFFFFFu);                  // global_addr[31:0]
  g0[3] = (unsigned)((ga >> 32) & 0x1FFFFFFu)            // global_addr[56:32]
          | (2u << 30);                                  // type = 2 ("image")

  v8i g1;
  g1[0] = (int)(2u << 16);        // workgroup_mask=0, data_size=2 (4 bytes)
  g1[1] = (int)(256u << 16);      // tensor_dim0 = 256 (bits 63:48)
  g1[2] = (int)(256u << 16);      // tensor_dim1 = 256 (bits 95:80)
  g1[3] = (int)(256u << 16);      // tile_dim0   = 256 (bits 127:112)
  g1[4] = (int)256;               // tile_dim1   = 256, tile_dim2 = 0
  g1[5] = (int)256;               // tensor_dim0_stride = 256
  g1[6] = 0;                      // stride0 hi / stride1 lo16 (65536 & 0xFFFF = 0)
  g1[7] = 1;                      // tensor_dim1_stride[47:16] = 65536 >> 16
#if __has_builtin(__builtin_amdgcn_tensor_load_to_lds_d2)
  __builtin_amdgcn_tensor_load_to_lds_d2(g0, g1, 0);
#else
  v4i z4 = {0, 0, 0, 0};
#if defined(__clang_major__) && (__clang_major__ >= 23)
  v8i z8 = {0, 0, 0, 0, 0, 0, 0, 0};
  __builtin_amdgcn_tensor_load_to_lds(g0, g1, z4, z4, z8, 0);
#else
  __builtin_amdgcn_tensor_load_to_lds(g0, g1, z4, z4, 0);
#endif
#endif
  tensor_wait0();
#else
  (void)gsrc; (void)ldst;
#endif
}

// =====================================================================
// Phase 1: forward max-plus DP. One block per batch element, one thread
// per "cur" tag. Transitions resident in LDS (256 KB), alphas ping-pong
// in LDS, emissions double-buffered via async-to-LDS.
// =====================================================================
__global__ __launch_bounds__(NTAGS, 1)
void viterbi_forward(const float* __restrict__ em,     // [B, T, N]
                     const float* __restrict__ mask,   // [B, T]
                     const float* __restrict__ trans,  // [N, N]
                     float* __restrict__ best_out,     // [B]
                     float* __restrict__ paths_out,    // [B, T]
                     unsigned char* __restrict__ bpt,  // [B, T-1, N]
                     int* __restrict__ final_tags)     // [B]
{
  __shared__ __align__(16) float trans_s[NTAGS * NTAGS];  // 256 KB
  __shared__ __align__(16) float alpha_s[2][NTAGS];       // ping-pong
  __shared__ __align__(16) float em_s[2][NTAGS];          // double buffer

  const int b   = blockIdx.x;
  const int tid = threadIdx.x;
  const float* emb   = em   + (size_t)b * SEQ_T * NTAGS;
  const float* maskb = mask + (size_t)b * SEQ_T;
  unsigned char* bptb = bpt + (size_t)b * (SEQ_T - 1) * NTAGS;

  // Prefetch emissions for t=0 (each lane fetches its own slot).
  async_cp_f32(&emb[tid], &em_s[0][tid]);

  // Bring transitions into LDS.
#if HAVE_TDM
  if (tid < 32) {                 // one wave issues the DMA
    tdm_load_trans(trans, trans_s);
  }
#else
  {
    const float4* src = reinterpret_cast<const float4*>(trans);
    float4* dst = reinterpret_cast<float4*>(trans_s);
    for (int i = tid; i < (NTAGS * NTAGS) / 4; i += NTAGS) dst[i] = src[i];
  }
#endif

  async_wait0();
  // alpha_0[cur] = trans[SOS][cur] + em[b,0,cur]
  alpha_s[0][tid] = trans_s[TAG_SOS * NTAGS + tid] + em_s[0][tid];
  __syncthreads();

  int p = 0;
  for (int t = 1; t < SEQ_T; ++t) {
    const int buf = t & 1;
    // Prefetch this step's emission slot asynchronously; consumed after
    // the (long) max-plus reduction below hides the latency.
    async_cp_f32(&emb[(size_t)t * NTAGS + tid], &em_s[buf][tid]);

    const float aold = alpha_s[p][tid];
    float best = -3.4e38f;
    int   arg  = 0;
    const float* __restrict__ ap = alpha_s[p];
#pragma unroll 4
    for (int pv = 0; pv < NTAGS; pv += 4) {
      const float4 av = *reinterpret_cast<const float4*>(ap + pv);  // LDS bcast
      const float s0 = av.x + trans_s[(pv + 0) * NTAGS + tid];
      const float s1 = av.y + trans_s[(pv + 1) * NTAGS + tid];
      const float s2 = av.z + trans_s[(pv + 2) * NTAGS + tid];
      const float s3 = av.w + trans_s[(pv + 3) * NTAGS + tid];
      if (s0 > best) { best = s0; arg = pv;     }
      if (s1 > best) { best = s1; arg = pv + 1; }
      if (s2 > best) { best = s2; arg = pv + 2; }
      if (s3 > best) { best = s3; arg = pv + 3; }
    }

    async_wait0();
    const float m   = maskb[t];                 // uniform -> scalar load
    const float emt = em_s[buf][tid];
    // new = m*(best+em) + (1-m)*aold  ==  aold + m*((best+em) - aold)
    const float anew = aold + m * ((best + emt) - aold);
    alpha_s[p ^ 1][tid] = anew;
    bptb[(size_t)(t - 1) * NTAGS + tid] = (unsigned char)arg;
    __syncthreads();
    p ^= 1;
  }

  // end_scores[cur] = alpha[cur] + trans[cur][EOS]
  const float endsc = alpha_s[p][tid] + trans_s[tid * NTAGS + TAG_EOS];
  em_s[0][tid] = endsc;           // reuse as reduction scratch
  __syncthreads();
  if (tid == 0) {
    float bmax = em_s[0][0];
    int   barg = 0;
    for (int i = 1; i < NTAGS; ++i) {
      const float v = em_s[0][i];
      if (v > bmax) { bmax = v; barg = i; }   // strict > => first argmax
    }
    best_out[b]   = bmax;
    final_tags[b] = barg;
    paths_out[(size_t)b * SEQ_T + (SEQ_T - 1)] = (float)barg;
  }
}

// =====================================================================
// Phase 2: backtrace. One thread per batch, dependent byte-load chain
// through the (L2-resident) backpointer table.
// =====================================================================
__global__ __launch_bounds__(32)
void viterbi_backtrace(const unsigned char* __restrict__ bpt,
                       const int* __restrict__ final_tags,
                       float* __restrict__ paths_out, int B)
{
  const int b = blockIdx.x * blockDim.x + threadIdx.x;
  if (b >= B) return;
  int tag = final_tags[b];
  const unsigned char* bb = bpt + (size_t)b * (SEQ_T - 1) * NTAGS;
  float* pb = paths_out + (size_t)b * SEQ_T;
  for (int t = SEQ_T - 2; t >= 0; --t) {
    tag = (int)bb[(size_t)t * NTAGS + tag];
    pb[t] = (float)tag;
  }
}

extern "C" void kernel_launch(void* const* d_in, const int* in_sizes, int n_in,
                              void* d_out, int out_size, void* d_ws, size_t ws_size,
                              hipStream_t stream) {
  const float* em    = (const float*)d_in[0];  // [B, T, N]
  const float* mask  = (const float*)d_in[1];  // [B, T]
  const float* trans = (const float*)d_in[2];  // [N, N]

  const int B = in_sizes[1] / SEQ_T;           // mask is [B, T]

  float* best_out  = (float*)d_out;            // [B]
  float* paths_out = (float*)d_out + B;        // [B, T]

  unsigned char* bpt = (unsigned char*)d_ws;                       // [B, T-1, N]
  const size_t bpt_bytes = (size_t)B * (SEQ_T - 1) * NTAGS;
  int* final_tags = (int*)((char*)d_ws + bpt_bytes);               // [B]
  (void)ws_size; (void)n_in; (void)out_size;

  viterbi_forward<<<dim3(B), dim3(NTAGS), 0, stream>>>(
      em, mask, trans, best_out, paths_out, bpt, final_tags);
  viterbi_backtrace<<<dim3((B + 31) / 32), dim3(32), 0, stream>>>(
      bpt, final_tags, paths_out, B);
}